// MultiScaleSIMHead_77953656422849
// MI455X (gfx1250) — compile-verified
//
#include <hip/hip_runtime.h>
#include <hip/hip_bf16.h>

typedef __attribute__((ext_vector_type(2))) float v2f;
typedef __attribute__((ext_vector_type(8))) float v8f;

#define C_DIM 768
#define TD    512
#define K_DIM 4
#define NCG   (C_DIM / 4)   // 192 channel groups of 4

// ---------------------------------------------------------------------------
// Kernel 1: tmp[k][c] = dot(text[k,:512], W[c,:512]) + b[c]
// one wave32 per (k,c) output, shuffle reduction
// ---------------------------------------------------------------------------
__global__ __launch_bounds__(256) void proj_kernel(const float* __restrict__ text,
                                                   const float* __restrict__ W,
                                                   const float* __restrict__ bias,
                                                   float* __restrict__ tmp) {
    int wave = (blockIdx.x * 256 + threadIdx.x) >> 5;
    int lane = threadIdx.x & 31;
    if (wave >= K_DIM * C_DIM) return;
    int k = wave / C_DIM;
    int c = wave - k * C_DIM;
    const float* t = text + k * TD;
    const float* w = W + (size_t)c * TD;
    float s = 0.f;
#pragma unroll 4
    for (int i = lane; i < TD; i += 32) s = fmaf(t[i], w[i], s);
    for (int off = 16; off > 0; off >>= 1) s += __shfl_xor(s, off);
    if (lane == 0) tmp[k * C_DIM + c] = s + bias[c];
}

__device__ __forceinline__ float block_reduce_sum(float v, float* red) {
    int tid = threadIdx.x;
    red[tid] = v;
    __syncthreads();
    for (int off = 128; off > 0; off >>= 1) {
        if (tid < off) red[tid] += red[tid + off];
        __syncthreads();
    }
    float r = red[0];
    __syncthreads();
    return r;
}

// ---------------------------------------------------------------------------
// Kernel 2: per-k layernorm then l2-normalize; fold in logit scale.
// grid = 4 (one block per k), block = 256, each thread owns 3 channels.
// ---------------------------------------------------------------------------
__global__ __launch_bounds__(256) void norm_kernel(const float* __restrict__ tmp,
                                                   const float* __restrict__ g,
                                                   const float* __restrict__ beta,
                                                   const float* __restrict__ logit_scale,
                                                   float* __restrict__ wout) {
    __shared__ float red[256];
    int k   = blockIdx.x;
    int tid = threadIdx.x;
    const float* x = tmp + k * C_DIM;
    float xs[3];
    float s = 0.f, s2 = 0.f;
#pragma unroll
    for (int i = 0; i < 3; ++i) {
        float v = x[tid + 256 * i];
        xs[i] = v;
        s  += v;
        s2 += v * v;
    }
    float sum  = block_reduce_sum(s,  red);
    float sum2 = block_reduce_sum(s2, red);
    float mean = sum * (1.f / C_DIM);
    float var  = sum2 * (1.f / C_DIM) - mean * mean;
    float inv  = 1.f / sqrtf(var + 1e-5f);
    float ys[3];
    float ss = 0.f;
#pragma unroll
    for (int i = 0; i < 3; ++i) {
        int c = tid + 256 * i;
        float y = (xs[i] - mean) * inv * g[c] + beta[c];
        ys[i] = y;
        ss += y * y;
    }
    float sumsq = block_reduce_sum(ss, red);
    float l2 = fmaxf(sqrtf(sumsq), 1e-12f);
    float sc = fminf(fmaxf(expf(logit_scale[0]), 1.f), 100.f);
    float m = sc / l2;
#pragma unroll
    for (int i = 0; i < 3; ++i) wout[k * C_DIM + tid + 256 * i] = ys[i] * m;
}

// ---------------------------------------------------------------------------
// Kernel 3: pack w (already scaled) into per-lane B fragments for
// V_WMMA_F32_16X16X4_F32.  B is 4x16 (K x N): lanes 0-15 hold rows K=0 (v0)
// and K=1 (v1) for column N=lane; lanes 16-31 hold rows K=2/K=3.
// Columns N>=4 (unused text classes) are zero.
// wB layout: [cg][lane] -> float2, cg = channel group (4 channels each).
// ---------------------------------------------------------------------------
__global__ __launch_bounds__(256) void buildB_kernel(const float* __restrict__ w,
                                                     float* __restrict__ wB) {
    int idx = blockIdx.x * 256 + threadIdx.x;   // 0 .. NCG*32-1
    if (idx >= NCG * 32) return;
    int cg   = idx >> 5;
    int lane = idx & 31;
    int half = lane >> 4;
    int n    = lane & 15;
    float v0 = 0.f, v1 = 0.f;
    if (n < K_DIM) {
        int c = cg * 4 + half * 2;
        v0 = w[n * C_DIM + c];
        v1 = w[n * C_DIM + c + 1];
    }
    wB[idx * 2]     = v0;
    wB[idx * 2 + 1] = v1;
}

// ---------------------------------------------------------------------------
// Kernel 4 (hot): one wave per 16-pixel tile.  Single pass over all 768
// channels: accumulate cam via v_wmma_f32_16x16x4_f32 and sum-of-squares via
// VALU, then apply 1/||f|| row scaling and write the CAM.
// HW is a template parameter so channel strides fold into immediate load
// offsets (no per-iteration 64-bit address math).  f loads are non-temporal:
// ~594 MB streamed exactly once must not evict L2-resident wB/cam.
// f layout: [B=16][C=768][HW] ; cam layout: [B][K=4][HW]
// ---------------------------------------------------------------------------
template <int HW>
__global__ __launch_bounds__(256) void sim_kernel(const float* __restrict__ f,
                                                  const float* __restrict__ wB,
                                                  float* __restrict__ cam) {
    int lane = threadIdx.x & 31;
    int wave = threadIdx.x >> 5;
    int tile = blockIdx.x * 8 + wave;          // grid sized exactly
    int pix0 = tile * 16;
    int b    = pix0 / HW;                      // tiles never cross images (HW%16==0)
    int hw   = pix0 - b * HW;
    int half = lane >> 4;
    int m    = lane & 15;

    // this lane's pixel column, at its half-wave's first channel
    const float* p   = f + (size_t)b * C_DIM * HW + hw + m + (size_t)(half * 2) * HW;
    const v2f*   wB2 = (const v2f*)wB + lane;

    v8f acc = {0.f, 0.f, 0.f, 0.f, 0.f, 0.f, 0.f, 0.f};
    float ss = 0.f;

#pragma unroll 8
    for (int cg = 0; cg < NCG; ++cg) {
        float a0 = __builtin_nontemporal_load(p);        // channel c
        float a1 = __builtin_nontemporal_load(p + HW);   // channel c+1
        p += 4 * HW;                                     // next channel group
        v2f afrag = {a0, a1};
        v2f bfrag = wB2[cg * 32];
        ss = fmaf(a0, a0, ss);
        ss = fmaf(a1, a1, ss);
        // D(16x16 f32) += A(16x4 f32) x B(4x16 f32)
        acc = __builtin_amdgcn_wmma_f32_16x16x4_f32(false, afrag, false, bfrag,
                                                    (short)0, acc, false, false);
    }

    // complete per-pixel sum of squares: pixel m's partials live in lanes m, m+16
    ss += __shfl_xor(ss, 16);
    float rn = 1.f / fmaxf(sqrtf(ss), 1e-12f); // lanes 0-15 hold rnorm[pixel=lane]

    // scale each D row (pixel) by its rnorm; row of acc[j] is pixel j (lo half)
    // or j+8 (hi half)
    float vals[8];
#pragma unroll
    for (int j = 0; j < 8; ++j) {
        float rlo = __shfl(rn, j);
        float rhi = __shfl(rn, j + 8);
        float r = half ? rhi : rlo;
        vals[j] = acc[j] * r;
    }

    // lanes with N<4 write cam[b][n][hw + pixel]; 8 consecutive floats each
    if (m < K_DIM) {
        float* cp = cam + ((size_t)(b * K_DIM + m)) * HW + hw + half * 8;
#pragma unroll
        for (int j = 0; j < 8; ++j) cp[j] = vals[j];
    }
}

// ---------------------------------------------------------------------------
// Kernel 5: GAP over (H,W) -> logits[b][k].  One block per (b,k) plane.
// Deterministic tree reduction (no float atomics).
// ---------------------------------------------------------------------------
__global__ __launch_bounds__(256) void gap_kernel(const float* __restrict__ cam,
                                                  float* __restrict__ out,
                                                  int HW) {
    __shared__ float red[256];
    int bk = blockIdx.x;                       // 0..63
    const float* p = cam + (size_t)bk * HW;
    float s = 0.f;
    for (int i = threadIdx.x; i < HW; i += 256) s += p[i];
    float sum = block_reduce_sum(s, red);
    if (threadIdx.x == 0) out[bk] = sum / (float)HW;
}

// ---------------------------------------------------------------------------
extern "C" void kernel_launch(void* const* d_in, const int* in_sizes, int n_in,
                              void* d_out, int out_size, void* d_ws, size_t ws_size,
                              hipStream_t stream) {
    (void)in_sizes; (void)n_in; (void)out_size; (void)ws_size;
    const float* f[3]    = {(const float*)d_in[0], (const float*)d_in[1], (const float*)d_in[2]};
    const float* text    = (const float*)d_in[3];
    const float* Wm[3]   = {(const float*)d_in[4],  (const float*)d_in[8],  (const float*)d_in[12]};
    const float* bm[3]   = {(const float*)d_in[5],  (const float*)d_in[9],  (const float*)d_in[13]};
    const float* gm[3]   = {(const float*)d_in[6],  (const float*)d_in[10], (const float*)d_in[14]};
    const float* betam[3]= {(const float*)d_in[7],  (const float*)d_in[11], (const float*)d_in[15]};
    const float* ls      = (const float*)d_in[16];

    float* out = (float*)d_out;
    // output layout: l2[64], l3[64], l4[64], c2[16*4*9216], c3[16*4*2304], c4[16*4*576]
    float* logits[3] = {out, out + 64, out + 128};
    float* cam[3];
    cam[0] = out + 192;
    cam[1] = cam[0] + (size_t)16 * K_DIM * 9216;
    cam[2] = cam[1] + (size_t)16 * K_DIM * 2304;
    const int HW[3] = {96 * 96, 48 * 48, 24 * 24};

    // workspace (reused across scales; stream-ordered): tmp[3072] w[3072] wB[NCG*32*2]
    float* tmp = (float*)d_ws;
    float* wv  = tmp + K_DIM * C_DIM;
    float* wB  = wv  + K_DIM * C_DIM;

    for (int s = 0; s < 3; ++s) {
        proj_kernel <<<(K_DIM * C_DIM) / 8, 256, 0, stream>>>(text, Wm[s], bm[s], tmp);
        norm_kernel <<<K_DIM,               256, 0, stream>>>(tmp, gm[s], betam[s], ls, wv);
        buildB_kernel<<<(NCG * 32) / 256,   256, 0, stream>>>(wv, wB);
        int nblocks = (16 * HW[s] / 16) / 8;   // (B*HW/16 tiles) / 8 waves per block
        switch (s) {
            case 0: sim_kernel<96 * 96><<<nblocks, 256, 0, stream>>>(f[s], wB, cam[s]); break;
            case 1: sim_kernel<48 * 48><<<nblocks, 256, 0, stream>>>(f[s], wB, cam[s]); break;
            case 2: sim_kernel<24 * 24><<<nblocks, 256, 0, stream>>>(f[s], wB, cam[s]); break;
        }
        gap_kernel  <<<16 * K_DIM,          256, 0, stream>>>(cam[s], logits[s], HW[s]);
    }
}